// ScatterND_26336739459712
// MI455X (gfx1250) — compile-verified
//
#include <hip/hip_runtime.h>

// ---------------------------------------------------------------------------
// ScatterND for MI455X (gfx1250): out = x; out[indices[m,0], indices[m,1], :] = updates[m, :]
// Pure data movement (~2.35 GB) -> HBM-bandwidth bound (~100us @ 23.3 TB/s).
// Kernel 1: guard-free NT b128 copy; block-contiguous 32KB tiles so all 8
//           loads/stores use one address register + immediate offsets.
// Kernel 1b: tail copy for non-multiple sizes (not launched for this shape).
// Kernel 2: 4 slices per wave32; async LDS-direct path, 4 outstanding ops.
// ---------------------------------------------------------------------------

typedef float float4v __attribute__((ext_vector_type(4)));
typedef int   int4v   __attribute__((ext_vector_type(4)));

// Builtin parameter types (from clang diagnostic): global int4*, LDS int4*.
typedef __attribute__((address_space(1))) int4v* g_int4p;
typedef __attribute__((address_space(3))) int4v* l_int4p;

#define HAS_ASYNC_LDS (__has_builtin(__builtin_amdgcn_global_load_async_to_lds_b128) && \
                       __has_builtin(__builtin_amdgcn_global_store_async_from_lds_b128) && \
                       __has_builtin(__builtin_amdgcn_s_wait_asynccnt))

#define COPY_UNROLL 8
#define SLICES_PER_WAVE 4

// ---------------- Kernel 1: bulk copy x -> out (non-temporal, b128) --------
// Each block owns a contiguous 256*8 float4 (32KB) tile. Thread t touches
// tile + t + k*256, so k*256 float4 = k*4096 bytes is a compile-time VMEM
// immediate offset: one address pair, 8 NT b128 loads + 8 NT b128 stores.
__global__ __launch_bounds__(256) void nt_copy_b128_main(const float4v* __restrict__ src,
                                                         float4v* __restrict__ dst) {
    const long long base =
        (long long)blockIdx.x * (256 * COPY_UNROLL) + threadIdx.x;

    float4v v[COPY_UNROLL];
#pragma unroll
    for (int k = 0; k < COPY_UNROLL; ++k)
        v[k] = __builtin_nontemporal_load(src + base + k * 256);
#pragma unroll
    for (int k = 0; k < COPY_UNROLL; ++k)
        __builtin_nontemporal_store(v[k], dst + base + k * 256);
}

// ---------------- Kernel 1b: tail (only if n4 % (256*8) != 0) --------------
__global__ __launch_bounds__(256) void nt_copy_b128_tail(const float4v* __restrict__ src,
                                                         float4v* __restrict__ dst,
                                                         long long start, long long n4) {
    long long i = start + (long long)blockIdx.x * 256 + threadIdx.x;
    if (i < n4)
        __builtin_nontemporal_store(__builtin_nontemporal_load(src + i), dst + i);
}

// ---------------- Kernel 2: scatter 512B slices ----------------------------
// One wave (32 lanes x b128 = 512B) per slice; SLICES_PER_WAVE slices batched
// per wave so each wave keeps 4 async transfers outstanding.
__global__ __launch_bounds__(256) void scatter_slices(const int* __restrict__ idx,
                                                      const float4v* __restrict__ upd4,
                                                      float4v* __restrict__ out4,
                                                      int M, int d1, int c4) {
#if HAS_ASYNC_LDS
    __shared__ int4v stage[SLICES_PER_WAVE][256];   // 16KB/block
#endif
    const int t    = threadIdx.x;
    const int lane = t & 31;
    const long long sbase =
        ((long long)blockIdx.x * (256 / 32) + (t >> 5)) * SLICES_PER_WAVE;

    // chunk loop: each j-iteration moves 32 float4 (512B) of every batched slice
    for (int j = lane; j < c4; j += 32) {
        long long dst_off[SLICES_PER_WAVE];
        bool      valid[SLICES_PER_WAVE];

#pragma unroll
        for (int b = 0; b < SLICES_PER_WAVE; ++b) {
            const long long s = sbase + b;
            valid[b] = (s < (long long)M);
            if (valid[b]) {
                const int row = idx[2 * s + 0];
                const int col = idx[2 * s + 1];
                dst_off[b] = ((long long)row * d1 + col) * (long long)c4 + j;
#if HAS_ASYNC_LDS
                __builtin_amdgcn_global_load_async_to_lds_b128(
                    (g_int4p)(upd4 + s * c4 + j), (l_int4p)&stage[b][t], 0, 0);
#endif
            }
        }

#if HAS_ASYNC_LDS
        __builtin_amdgcn_s_wait_asynccnt(0);   // all batched LDS slots filled
#pragma unroll
        for (int b = 0; b < SLICES_PER_WAVE; ++b) {
            if (valid[b])
                __builtin_amdgcn_global_store_async_from_lds_b128(
                    (g_int4p)(out4 + dst_off[b]), (l_int4p)&stage[b][t], 0, 0);
        }
        __builtin_amdgcn_s_wait_asynccnt(0);   // WAR guard on slots before next j
#else
#pragma unroll
        for (int b = 0; b < SLICES_PER_WAVE; ++b) {
            if (valid[b]) {
                const long long s = sbase + b;
                __builtin_nontemporal_store(
                    __builtin_nontemporal_load(upd4 + s * c4 + j),
                    out4 + dst_off[b]);
            }
        }
#endif
    }
}

// ---------------------------------------------------------------------------
extern "C" void kernel_launch(void* const* d_in, const int* in_sizes, int n_in,
                              void* d_out, int out_size, void* d_ws, size_t ws_size,
                              hipStream_t stream) {
    const float* x       = (const float*)d_in[0];
    const int*   indices = (const int*)d_in[1];
    const float* updates = (const float*)d_in[2];
    float*       out     = (float*)d_out;

    const long long n  = (long long)in_sizes[0];     // D0*D1*C = 268435456
    const long long n4 = n >> 2;                     // float4 count = 67108864
    const int M  = in_sizes[1] / 2;                  // 200000 slices
    const int C  = (M > 0) ? in_sizes[2] / M : 128;  // 128
    const int c4 = C / 4;                            // 32 float4 per slice
    const int D1 = 2048;                             // from reference shape

    // ---- copy x -> out: exact-coverage main grid + optional tail ----------
    const long long per_block = 256LL * COPY_UNROLL;            // 2048 float4
    const long long main_blocks = n4 / per_block;               // 32768 exact
    if (main_blocks > 0)
        nt_copy_b128_main<<<(int)main_blocks, 256, 0, stream>>>(
            (const float4v*)x, (float4v*)out);
    const long long covered = main_blocks * per_block;
    const long long tail    = n4 - covered;                     // 0 here
    if (tail > 0) {
        int tb = (int)((tail + 255) / 256);
        nt_copy_b128_tail<<<tb, 256, 0, stream>>>(
            (const float4v*)x, (float4v*)out, covered, n4);
    }

    // ---- scatter: 8 waves/block, 4 slices/wave ----------------------------
    const int slices_per_block = (256 / 32) * SLICES_PER_WAVE;  // 32
    int sc_blocks = (M + slices_per_block - 1) / slices_per_block;  // 6250
    scatter_slices<<<sc_blocks, 256, 0, stream>>>(indices,
                                                  (const float4v*)updates,
                                                  (float4v*)out,
                                                  M, D1, c4);
}